// SphericalHarmonicsGaussianKernels_65309272703430
// MI455X (gfx1250) — compile-verified
//
#include <hip/hip_runtime.h>
#include <math.h>

// ---------------------------------------------------------------------------
// Compile-time construction of Y_MAT (16 SH rows x 20 monomials), mirroring
// the Python _build() (monomial basis sorted lexicographically, _P_poly,
// _A_poly, _B_poly, _pmul, normalization constants).
// ---------------------------------------------------------------------------
namespace shb {

constexpr double PI_ = 3.14159265358979323846264338327950288;

// lexicographically sorted (i,j,k) with i+j+k<=3  (matches Python sorted(set))
constexpr int MEXP[20][3] = {
    {0,0,0},{0,0,1},{0,0,2},{0,0,3},{0,1,0},{0,1,1},{0,1,2},{0,2,0},{0,2,1},{0,3,0},
    {1,0,0},{1,0,1},{1,0,2},{1,1,0},{1,1,1},{1,2,0},{2,0,0},{2,0,1},{2,1,0},{3,0,0}};

constexpr int monoPos(int i, int j, int k) {
    for (int t = 0; t < 20; ++t)
        if (MEXP[t][0] == i && MEXP[t][1] == j && MEXP[t][2] == k) return t;
    return 0;  // unreachable for degree <= 3
}
constexpr double fact(int n) { double r = 1.0; for (int i = 2; i <= n; ++i) r *= i; return r; }
constexpr double combd(int n, int k) { return (k < 0 || k > n) ? 0.0 : fact(n) / (fact(k) * fact(n - k)); }
constexpr double csqrt(double x) {
    double r = (x > 1.0) ? x : 1.0;
    for (int i = 0; i < 100; ++i) { double nr = 0.5 * (r + x / r); if (nr == r) break; r = nr; }
    return r;
}

struct Poly { double c[20]; };

constexpr Poly pmul(const Poly& p, const Poly& q) {
    Poly r{};
    for (int a = 0; a < 20; ++a) if (p.c[a] != 0.0)
        for (int b = 0; b < 20; ++b) if (q.c[b] != 0.0) {
            int i = MEXP[a][0] + MEXP[b][0];
            int j = MEXP[a][1] + MEXP[b][1];
            int k = MEXP[a][2] + MEXP[b][2];
            r.c[monoPos(i, j, k)] += p.c[a] * q.c[b];
        }
    return r;
}
constexpr Poly A_poly(int m) {
    Poly p{};
    const double cos4[4] = {1.0, 0.0, -1.0, 0.0};
    for (int q = 0; q <= m; ++q) {
        double cc = combd(m, q) * cos4[(m - q) % 4];
        if (cc != 0.0) p.c[monoPos(q, m - q, 0)] += cc;
    }
    return p;
}
constexpr Poly B_poly(int m) {
    Poly p{};
    const double sin4[4] = {0.0, 1.0, 0.0, -1.0};
    for (int q = 0; q <= m; ++q) {
        double cc = combd(m, q) * sin4[(m - q) % 4];
        if (cc != 0.0) p.c[monoPos(q, m - q, 0)] += cc;
    }
    return p;
}
constexpr Poly P_poly(int l, int m) {
    Poly p{};
    for (int k = 0; k <= (l - m) / 2; ++k) {
        double pk = ((k & 1) ? -1.0 : 1.0) * (1.0 / double(1 << l)) * combd(l, k) * combd(2 * l - 2 * k, l);
        pk *= fact(l - 2 * k) / fact(l - 2 * k - m);
        int ze = l - 2 * k - m;
        for (int a = 0; a <= k; ++a)
            for (int b = 0; b <= k - a; ++b) {
                int c = k - a - b;
                double coef = pk * fact(k) / (fact(a) * fact(b) * fact(c));
                p.c[monoPos(2 * a, 2 * b, 2 * c + ze)] += coef;
            }
    }
    double s = csqrt(fact(l - m) / fact(l + m));
    Poly r{};
    for (int t = 0; t < 20; ++t) r.c[t] = p.c[t] * s;
    return r;
}

struct YTab { float y[16][20]; };

constexpr YTab makeY() {
    YTab out{};
    int row = 0;
    for (int l = 0; l <= 3; ++l)
        for (int m = -l; m <= l; ++m) {
            Poly poly{}; double K = 0.0;
            if (m > 0)      { poly = pmul(P_poly(l, m),  A_poly(m));  K = csqrt((2.0 * l + 1.0) / (2.0 * PI_)); }
            else if (m < 0) { poly = pmul(P_poly(l, -m), B_poly(-m)); K = csqrt((2.0 * l + 1.0) / (2.0 * PI_)); }
            else            { poly = P_poly(l, 0);                    K = csqrt((2.0 * l + 1.0) / (4.0 * PI_)); }
            for (int t = 0; t < 20; ++t) out.y[row][t] = float(K * poly.c[t]);
            ++row;
        }
    return out;
}

}  // namespace shb

__constant__ shb::YTab c_Y = shb::makeY();

// ---------------------------------------------------------------------------
// Kernels
// ---------------------------------------------------------------------------
typedef float v2f __attribute__((ext_vector_type(2)));
typedef float v4f __attribute__((ext_vector_type(4)));
typedef float v8f __attribute__((ext_vector_type(8)));

#define NB 16
#define NV 2048
#define NP 32
#define NSH 16
#define NR 3
#define NMONO 20
#define WPB 8  // waves (== vertices) per 256-thread block

__device__ __forceinline__ int sh_l(int s) { return (s == 0) ? 0 : (s < 4) ? 1 : (s < 9) ? 2 : 3; }

// Per-wave: sh(16 x 16patches) = Y(16x20) * mono(20x16) via 5 chained
// V_WMMA_F32_16X16X4_F32. monoLds points at mono[patch0..15][20] for this group.
// f32 A 16x4 layout: lane L -> M=L&15, VGPR0 = A[M][2*(L>>4)], VGPR1 = +1 (per chunk).
// f32 B 4x16 layout: lane L -> N=L&15, VGPR0 = B[2*(L>>4)][N], VGPR1 = +1.
__device__ __forceinline__ v8f wave_sh_wmma(const float* monoLds) {
    const int lane = threadIdx.x & 31;
    const int m16 = lane & 15;
    const int hi = lane >> 4;
    v8f acc = {};
#pragma unroll
    for (int c = 0; c < 5; ++c) {
        const int k0 = 4 * c + 2 * hi;
        v2f a, b;
        a.x = c_Y.y[m16][k0];
        a.y = c_Y.y[m16][k0 + 1];
        b.x = monoLds[m16 * NMONO + k0];
        b.y = monoLds[m16 * NMONO + k0 + 1];
        acc = __builtin_amdgcn_wmma_f32_16x16x4_f32(false, a, false, b, (short)0, acc, false, false);
    }
    return acc;
}

// Shared geometry computation: returns dist and writes monomials into mo[20].
__device__ __forceinline__ float compute_mono(const float* __restrict__ pp, float* mo) {
    float x = pp[0], y = pp[1], z = pp[2];
    float dist = sqrtf(x * x + y * y + z * z);
    float inv = 1.0f / fmaxf(dist, 1e-6f);
    float nx = -x * inv, ny = -y * inv, nz = -z * inv;
    float x2 = nx * nx, y2 = ny * ny, z2 = nz * nz;
    mo[0] = 1.0f;      mo[1] = nz;        mo[2] = z2;        mo[3] = z2 * nz;
    mo[4] = ny;        mo[5] = ny * nz;   mo[6] = ny * z2;   mo[7] = y2;
    mo[8] = y2 * nz;   mo[9] = y2 * ny;   mo[10] = nx;       mo[11] = nx * nz;
    mo[12] = nx * z2;  mo[13] = nx * ny;  mo[14] = nx * ny * nz; mo[15] = nx * y2;
    mo[16] = x2;       mo[17] = x2 * nz;  mo[18] = x2 * ny;  mo[19] = x2 * nx;
    return dist;
}

__device__ __forceinline__ void compute_shells(float dist, float& s0, float& s1, float& s2) {
    float d0 = dist, d1 = dist - 0.5f, d2 = dist - 1.0f;
    float e0 = expf(-6.25f * d0 * d0);
    float e1 = expf(-6.25f * d1 * d1);
    float e2 = expf(-6.25f * d2 * d2);
    float ssum = fmaxf(e0 + e1 + e2, 1e-6f);
    s0 = e0 / ssum; s1 = e1 / ssum; s2 = e2 / ssum;
    if (dist > 1.0f) { s0 = 0.0f; s1 = 0.0f; s2 = 0.0f; }
}

// Pass 1: per-vertex l2 over P (done on the matrix unit), group by l, sqrt,
// accumulate sum over V per batch.
__global__ void __launch_bounds__(256) sh_pass1(const float* __restrict__ patches,
                                                float* __restrict__ accum) {
    __shared__ float sMono[WPB][NP][NMONO];
    __shared__ float sW4[WPB][NP][4];     // shells^2 (bound-masked), [3] == 0 pad
    __shared__ float sSh2[WPB][NP][NSH];  // sh^2
    __shared__ float sL2[WPB][NSH][NR];
    __shared__ float sMl[12];
    const int tid = threadIdx.x;
    const int w = tid >> 5;
    const int lane = tid & 31;
    const int m16 = lane & 15;
    const int hi = lane >> 4;
    const size_t vert = (size_t)blockIdx.x * WPB + w;
    const int b = (int)(vert >> 11);  // V == 2048

    if (tid < 12) sMl[tid] = 0.0f;

    float mo[NMONO];
    float dist = compute_mono(patches + (vert * NP + lane) * 3, mo);
#pragma unroll
    for (int k = 0; k < NMONO; ++k) sMono[w][lane][k] = mo[k];

    float s0, s1, s2;
    compute_shells(dist, s0, s1, s2);
    {
        v4f wq = {s0 * s0, s1 * s1, s2 * s2, 0.0f};
        *reinterpret_cast<v4f*>(&sW4[w][lane][0]) = wq;
    }

    __syncthreads();

    // Stage 1: sh = Y * mono (WMMA), square, stage sh^2 to LDS.
    // Lane holds sh[s = j+8*hi] for patch p = g*16+m16; the 8 s-values are
    // contiguous -> two 16B LDS stores per group.
#pragma unroll
    for (int g = 0; g < 2; ++g) {
        v8f sh = wave_sh_wmma(&sMono[w][g * 16][0]);
        int p = g * 16 + m16;
        v4f q0 = {sh[0]*sh[0], sh[1]*sh[1], sh[2]*sh[2], sh[3]*sh[3]};
        v4f q1 = {sh[4]*sh[4], sh[5]*sh[5], sh[6]*sh[6], sh[7]*sh[7]};
        v4f* dst = reinterpret_cast<v4f*>(&sSh2[w][p][hi * 8]);
        dst[0] = q0;
        dst[1] = q1;
    }
    __syncthreads();

    // Stage 2: l2[r][s] = sum_p w[p][r] * sh2[p][s]  ==  W(3x32,pad16) * SH2(32x16)
    // as 8 chained K=4 f32 WMMAs. A[M=r][K=p] = w[p][r] (rows 3..15 zero via the
    // zero-padded column 3 + clamped index -> unconditional LDS loads),
    // B[K=p][N=s] = sh2[p][s].
    {
        const int rm = (m16 < 3) ? m16 : 3;  // sW4[..][3] == 0 -> zero rows
        v8f acc = {};
#pragma unroll
        for (int c = 0; c < 8; ++c) {
            const int k0 = 4 * c + 2 * hi;
            v2f a, bv;
            a.x = sW4[w][k0][rm];
            a.y = sW4[w][k0 + 1][rm];
            bv.x = sSh2[w][k0][m16];
            bv.y = sSh2[w][k0 + 1][m16];
            acc = __builtin_amdgcn_wmma_f32_16x16x4_f32(false, a, false, bv, (short)0, acc, false, false);
        }
        // D layout: lane (hi==0, m16=s) holds l2[r=j][s] in acc[j], j=0..2.
        if (hi == 0) {
            sL2[w][m16][0] = acc[0];
            sL2[w][m16][1] = acc[1];
            sL2[w][m16][2] = acc[2];
        }
    }
    __syncthreads();

    if (lane < 12) {
        int l = lane / 3, r = lane - l * 3;
        // SH_IDX groups are contiguous: l=0 -> [0,1), l=1 -> [1,4), l=2 -> [4,9), l=3 -> [9,16)
        const int start = (l == 0) ? 0 : (l == 1) ? 1 : (l == 2) ? 4 : 9;
        const int stop  = (l == 0) ? 1 : (l == 1) ? 4 : (l == 2) ? 9 : 16;
        float sum = 0.0f;
        for (int s = start; s < stop; ++s) sum += sL2[w][s][r];
        float ml = sqrtf(sum + 1e-7f);
        atomicAdd(&sMl[lane], ml);
    }
    __syncthreads();
    if (tid < 12) atomicAdd(&accum[b * 12 + tid], sMl[tid]);
}

// Pass 2: recompute sh, divide by per-batch norm, write output coalesced (NT).
__global__ void __launch_bounds__(256) sh_pass2(const float* __restrict__ patches,
                                                const float* __restrict__ accum,
                                                float* __restrict__ out) {
    __shared__ float sMono[WPB][NP][NMONO];
    __shared__ float sSh[WPB][NP][NSH];
    __shared__ float sShell[WPB][NP][NR];
    __shared__ float sInvN[NSH][NR];
    const int tid = threadIdx.x;
    const int w = tid >> 5;
    const int lane = tid & 31;
    const int m16 = lane & 15;
    const int hi = lane >> 4;
    const size_t vert = (size_t)blockIdx.x * WPB + w;
    const int b = (int)(vert >> 11);

    if (tid < NSH * NR) {
        int s = tid / 3, r = tid - s * 3;
        int l = sh_l(s);
        float nm = fmaxf(accum[b * 12 + l * 3 + r] * (1.0f / (float)NV), 1e-8f) + 1e-6f;
        sInvN[s][r] = 1.0f / nm;
    }

    float mo[NMONO];
    float dist = compute_mono(patches + (vert * NP + lane) * 3, mo);
#pragma unroll
    for (int k = 0; k < NMONO; ++k) sMono[w][lane][k] = mo[k];

    float s0, s1, s2;
    compute_shells(dist, s0, s1, s2);
    sShell[w][lane][0] = s0; sShell[w][lane][1] = s1; sShell[w][lane][2] = s2;

    __syncthreads();

#pragma unroll
    for (int g = 0; g < 2; ++g) {
        v8f sh = wave_sh_wmma(&sMono[w][g * 16][0]);
        int p = g * 16 + m16;
        v4f q0 = {sh[0], sh[1], sh[2], sh[3]};
        v4f q1 = {sh[4], sh[5], sh[6], sh[7]};
        v4f* dst = reinterpret_cast<v4f*>(&sSh[w][p][hi * 8]);
        dst[0] = q0;
        dst[1] = q1;
    }
    __syncthreads();

    // 1536 contiguous floats per vertex; wave writes them as 384 16B stores,
    // non-temporal (output is 200MB > 192MB L2, don't thrash the cache).
    float* po = out + vert * (size_t)(NP * NSH * NR);
#pragma unroll
    for (int i = 0; i < 12; ++i) {
        int e = (i * 32 + lane) * 4;  // 48 % 4 == 0 -> one patch per 16B store
        int p = e / 48;
        int rem = e - p * 48;
        float vv[4];
#pragma unroll
        for (int q = 0; q < 4; ++q) {
            int s = (rem + q) / 3, r = (rem + q) - s * 3;
            vv[q] = sSh[w][p][s] * sShell[w][p][r] * sInvN[s][r];
        }
        v4f vr = {vv[0], vv[1], vv[2], vv[3]};
        __builtin_nontemporal_store(vr, reinterpret_cast<v4f*>(po + e));
    }
}

// ---------------------------------------------------------------------------
extern "C" void kernel_launch(void* const* d_in, const int* in_sizes, int n_in,
                              void* d_out, int out_size, void* d_ws, size_t ws_size,
                              hipStream_t stream) {
    const float* patches = (const float*)d_in[0];
    float* out = (float*)d_out;
    float* accum = (float*)d_ws;  // 16 batches * 4 l * 3 r = 192 floats

    (void)hipMemsetAsync(accum, 0, NB * 12 * sizeof(float), stream);

    dim3 block(256);
    dim3 grid((NB * NV) / WPB);  // 4096 blocks, 8 vertices (waves) each
    sh_pass1<<<grid, block, 0, stream>>>(patches, accum);
    sh_pass2<<<grid, block, 0, stream>>>(patches, accum, out);
}